// Bert_BiLSTM_CRF_21251498180658
// MI455X (gfx1250) — compile-verified
//
#include <hip/hip_runtime.h>
#include <math.h>

#define SEQ   16384
#define DIN   768
#define HID   128
#define G4H   512   // 4*HID
#define NTAG  8

typedef __attribute__((ext_vector_type(2))) float v2f;
typedef __attribute__((ext_vector_type(8))) float v8f;

#if __has_builtin(__builtin_amdgcn_wmma_f32_16x16x4_f32)
#define HAVE_WMMA_F32 1
#else
#define HAVE_WMMA_F32 0
#endif

// ---------------------------------------------------------------------------
// Kernel 1: xp = x @ w^T + b for both directions (blockIdx.z selects dir).
// One wave (32 threads) per block; each wave owns a 16(M) x 64(N) output tile
// as 4 WMMA accumulators, reusing the A fragment across the 4 N-tiles.
// x: [SEQ x K] row-major, w: [512 x K] row-major, out: [SEQ x 512].
// ---------------------------------------------------------------------------
__global__ __launch_bounds__(32) void xproj_wmma(
    const float* __restrict__ x, int K,
    const float* __restrict__ wF, const float* __restrict__ bF, float* __restrict__ outF,
    const float* __restrict__ wB, const float* __restrict__ bB, float* __restrict__ outB)
{
    const float* w    = blockIdx.z ? wB   : wF;
    const float* bias = blockIdx.z ? bB   : bF;
    float*       out  = blockIdx.z ? outB : outF;

    const int lane = threadIdx.x;      // 0..31
    const int half = lane >> 4;        // 0: K=0,1  1: K=2,3 (A frag halves)
    const int l16  = lane & 15;
    const int row0 = blockIdx.x * 16;  // M tile over SEQ
    const int col0 = blockIdx.y * 64;  // 4 N-tiles of 16 over 512 gates

    v8f acc0 = {}, acc1 = {}, acc2 = {}, acc3 = {};

#if HAVE_WMMA_F32
    // A fragment: lane l<16 holds A[row0+l][k], A[row0+l][k+1];
    //             lane l>=16 holds A[row0+l-16][k+2], [k+3].
    const float* arow = x + (size_t)(row0 + l16) * K + 2 * half;
    // B fragment (K x 16): lane holds w[col0+t*16+l16][k+2*half .. +1]
    const float* br0 = w + (size_t)(col0 +  0 + l16) * K + 2 * half;
    const float* br1 = w + (size_t)(col0 + 16 + l16) * K + 2 * half;
    const float* br2 = w + (size_t)(col0 + 32 + l16) * K + 2 * half;
    const float* br3 = w + (size_t)(col0 + 48 + l16) * K + 2 * half;

    for (int k = 0; k < K; k += 4) {
        v2f a  = *(const v2f*)(arow + k);
        v2f b0 = *(const v2f*)(br0 + k);
        v2f b1 = *(const v2f*)(br1 + k);
        v2f b2 = *(const v2f*)(br2 + k);
        v2f b3 = *(const v2f*)(br3 + k);
        acc0 = __builtin_amdgcn_wmma_f32_16x16x4_f32(false, a, false, b0, (short)0, acc0, false, false);
        acc1 = __builtin_amdgcn_wmma_f32_16x16x4_f32(false, a, false, b1, (short)0, acc1, false, false);
        acc2 = __builtin_amdgcn_wmma_f32_16x16x4_f32(false, a, false, b2, (short)0, acc2, false, false);
        acc3 = __builtin_amdgcn_wmma_f32_16x16x4_f32(false, a, false, b3, (short)0, acc3, false, false);
    }
#else
    // Scalar fallback matching the WMMA C/D ownership (m = r+8*half, n = l16).
    for (int k = 0; k < K; ++k) {
        #pragma unroll
        for (int r = 0; r < 8; ++r) {
            float av = x[(size_t)(row0 + r + 8 * half) * K + k];
            acc0[r] += av * w[(size_t)(col0 +  0 + l16) * K + k];
            acc1[r] += av * w[(size_t)(col0 + 16 + l16) * K + k];
            acc2[r] += av * w[(size_t)(col0 + 32 + l16) * K + k];
            acc3[r] += av * w[(size_t)(col0 + 48 + l16) * K + k];
        }
    }
#endif

    // C/D layout: VGPR r -> M = r + 8*half (rows), N = l16 (cols).
    #pragma unroll
    for (int r = 0; r < 8; ++r) {
        int m = row0 + r + 8 * half;
        float* orow = out + (size_t)m * G4H + col0 + l16;
        orow[ 0] = acc0[r] + bias[col0 +  0 + l16];
        orow[16] = acc1[r] + bias[col0 + 16 + l16];
        orow[32] = acc2[r] + bias[col0 + 32 + l16];
        orow[48] = acc3[r] + bias[col0 + 48 + l16];
    }
}

// ---------------------------------------------------------------------------
// Kernel 2: persistent-workgroup LSTM recurrence, one block per direction.
// 512 threads: thread j owns gate j (row j of whh, held in 128 registers).
// h broadcast from LDS; c lives in registers of lanes 0..127.
// hout: [SEQ x 256], fwd -> cols 0..127, bwd -> cols 128..255.
// ---------------------------------------------------------------------------
__global__ __launch_bounds__(512) void lstm_layer(
    const float* __restrict__ xpF, const float* __restrict__ whhF,
    const float* __restrict__ xpB, const float* __restrict__ whhB,
    float* __restrict__ hout)
{
    __shared__ float sh_h[HID];
    __shared__ float sh_g[G4H];

    const int dir = blockIdx.x;                  // 0 = forward, 1 = backward
    const float* xp  = dir ? xpB  : xpF;
    const float* whh = dir ? whhB : whhF;
    const int j = threadIdx.x;                   // gate index 0..511

    // Hoist this thread's weight row (128 f32) into registers.
    float4 wreg[32];
    const float4* wr = (const float4*)(whh + (size_t)j * HID);
    #pragma unroll
    for (int k = 0; k < 32; ++k) wreg[k] = wr[k];

    float c = 0.0f;
    if (j < HID) sh_h[j] = 0.0f;
    __syncthreads();

    for (int step = 0; step < SEQ; ++step) {
        const int s = dir ? (SEQ - 1 - step) : step;

        float acc = xp[(size_t)s * G4H + j];
        const float4* hv = (const float4*)sh_h;
        #pragma unroll
        for (int k = 0; k < 32; ++k) {
            float4 h4 = hv[k];
            acc += wreg[k].x * h4.x + wreg[k].y * h4.y +
                   wreg[k].z * h4.z + wreg[k].w * h4.w;
        }
        sh_g[j] = acc;
        __syncthreads();

        if (j < HID) {
            float gi = sh_g[j];
            float gf = sh_g[j + 128];
            float gg = sh_g[j + 256];
            float go = sh_g[j + 384];
            float i_ = 1.0f / (1.0f + __expf(-gi));
            float f_ = 1.0f / (1.0f + __expf(-gf));
            float o_ = 1.0f / (1.0f + __expf(-go));
            c = f_ * c + i_ * tanhf(gg);
            float h = o_ * tanhf(c);
            sh_h[j] = h;
            hout[(size_t)s * (2 * HID) + dir * HID + j] = h;
        }
        __syncthreads();
    }
}

// ---------------------------------------------------------------------------
// Kernel 3: logits + CRF loss partials. One thread per sentence position.
// ---------------------------------------------------------------------------
__global__ __launch_bounds__(256) void crf_partial(
    const float* __restrict__ h1, const float* __restrict__ wlin,
    const float* __restrict__ blin, const float* __restrict__ stt,
    const float* __restrict__ ent, const int* __restrict__ tags,
    float* __restrict__ partial)
{
    __shared__ float swl[NTAG * 256];
    __shared__ float red[256];

    for (int i = threadIdx.x; i < NTAG * 256; i += 256) swl[i] = wlin[i];
    __syncthreads();

    const int s = blockIdx.x * 256 + threadIdx.x;
    const float4* hr = (const float4*)(h1 + (size_t)s * 256);

    float m[NTAG];
    #pragma unroll
    for (int t = 0; t < NTAG; ++t) {
        const float4* wv = (const float4*)(swl + t * 256);
        float acc = blin[t];
        #pragma unroll
        for (int k = 0; k < 64; ++k) {
            float4 h4 = hr[k];
            float4 w4 = wv[k];
            acc += h4.x * w4.x + h4.y * w4.y + h4.z * w4.z + h4.w * w4.w;
        }
        m[t] = acc + stt[t] + ent[t];
    }

    float mx = m[0];
    #pragma unroll
    for (int t = 1; t < NTAG; ++t) mx = fmaxf(mx, m[t]);
    float se = 0.0f;
    #pragma unroll
    for (int t = 0; t < NTAG; ++t) se += __expf(m[t] - mx);
    float logZ = mx + __logf(se);

    int tg = tags[s];
    float term = logZ - m[tg & (NTAG - 1)];   // loss contribution = logZ - score

    red[threadIdx.x] = term;
    __syncthreads();
    for (int off = 128; off > 0; off >>= 1) {
        if (threadIdx.x < off) red[threadIdx.x] += red[threadIdx.x + off];
        __syncthreads();
    }
    if (threadIdx.x == 0) partial[blockIdx.x] = red[0];
}

__global__ __launch_bounds__(64) void crf_final(const float* __restrict__ partial,
                                                float* __restrict__ out)
{
    __shared__ float red[64];
    red[threadIdx.x] = partial[threadIdx.x];
    __syncthreads();
    for (int off = 32; off > 0; off >>= 1) {
        if (threadIdx.x < off) red[threadIdx.x] += red[threadIdx.x + off];
        __syncthreads();
    }
    if (threadIdx.x == 0) out[0] = red[0] / (float)SEQ;
}

// ---------------------------------------------------------------------------
extern "C" void kernel_launch(void* const* d_in, const int* in_sizes, int n_in,
                              void* d_out, int out_size, void* d_ws, size_t ws_size,
                              hipStream_t stream)
{
    (void)in_sizes; (void)n_in; (void)out_size; (void)ws_size;

    const float* embeds = (const float*)d_in[0];
    const int*   tags   = (const int*)  d_in[1];
    const float* wi0f = (const float*)d_in[2];
    const float* wh0f = (const float*)d_in[3];
    const float* b0f  = (const float*)d_in[4];
    const float* wi0b = (const float*)d_in[5];
    const float* wh0b = (const float*)d_in[6];
    const float* b0b  = (const float*)d_in[7];
    const float* wi1f = (const float*)d_in[8];
    const float* wh1f = (const float*)d_in[9];
    const float* b1f  = (const float*)d_in[10];
    const float* wi1b = (const float*)d_in[11];
    const float* wh1b = (const float*)d_in[12];
    const float* b1b  = (const float*)d_in[13];
    const float* wlin = (const float*)d_in[14];
    const float* blin = (const float*)d_in[15];
    const float* stt  = (const float*)d_in[16];
    const float* ent  = (const float*)d_in[17];

    // Workspace layout (floats). xp buffers reused between layers.
    float* ws  = (float*)d_ws;
    float* xpF = ws;                                   // SEQ*512
    float* xpB = ws + (size_t)SEQ * G4H;               // SEQ*512
    float* h0  = ws + (size_t)2 * SEQ * G4H;           // SEQ*256
    float* h1  = h0 + (size_t)SEQ * 256;               // SEQ*256
    float* prt = h1 + (size_t)SEQ * 256;               // 64 partials

    dim3 gproj(SEQ / 16, G4H / 64, 2);

    // Layer 0: input projection (WMMA GEMM) then recurrence.
    xproj_wmma<<<gproj, 32, 0, stream>>>(embeds, DIN, wi0f, b0f, xpF, wi0b, b0b, xpB);
    lstm_layer<<<2, 512, 0, stream>>>(xpF, wh0f, xpB, wh0b, h0);

    // Layer 1: projection of concat(hf, hb) then recurrence.
    xproj_wmma<<<gproj, 32, 0, stream>>>(h0, 2 * HID, wi1f, b1f, xpF, wi1b, b1b, xpB);
    lstm_layer<<<2, 512, 0, stream>>>(xpF, wh1f, xpB, wh1b, h1);

    // Logits + CRF loss reduction.
    crf_partial<<<SEQ / 256, 256, 0, stream>>>(h1, wlin, blin, stt, ent, tags, prt);
    crf_final<<<1, 64, 0, stream>>>(prt, (float*)d_out);
}